// Graph_Attention_Union_76879914598932
// MI455X (gfx1250) — compile-verified
//
#include <hip/hip_runtime.h>
#include <math.h>

typedef __attribute__((ext_vector_type(2))) float v2f;
typedef __attribute__((ext_vector_type(8))) float v8f;

__device__ __forceinline__ v8f wmma4(v2f a, v2f b, v8f c) {
    // D = A(16x4) * B(4x16) + C(16x16), fp32 end-to-end (matches reference numerics)
    return __builtin_amdgcn_wmma_f32_16x16x4_f32(false, a, false, b, (short)0, c, false, false);
}

// ---------------------------------------------------------------------------
// Generic strided batched GEMM:  Y[b] = epilogue( A * B[b] )
//   A[m,k]   at A + b*sAb + m*sAm + k*sAk          (sAb=0 => shared weights)
//   B[k,n]   at (k<K0 ? B0 : B1 shifted) + b*sBb + k*sBk + n*sBn   (fused concat)
//   Y[b,m,n] at Y + b*sYb + m*ldY + n
//   epilogue: y = acc*scale[m] + shift[m]  (scale==null -> 1, shift==null -> 0), opt ReLU
// Block: 128 threads (4 waves), 64x64 tile, K staged 16 deep in LDS.
// Global->register->LDS software pipeline: the fetch for tile k+1 issues as an
// unpredicated 16-load clause that overlaps the 16 WMMAs of tile k.
// ---------------------------------------------------------------------------
__global__ __launch_bounds__(128) void wmma_gemm_f32(
    const float* __restrict__ A, long long sAm, long long sAk, long long sAb,
    const float* __restrict__ B0, const float* __restrict__ B1, int K0,
    long long sBk, long long sBn, long long sBb,
    float* __restrict__ Y, int ldY, long long sYb,
    int M, int N, int K,
    const float* __restrict__ scale, const float* __restrict__ shift, int relu)
{
    __shared__ float Al[64 * 18];   // [m][k], padded: even indices -> b64-aligned
    __shared__ float Bl[64 * 18];   // transposed [n][k], padded

    const int tid  = threadIdx.x;
    const int lane = tid & 31;
    const int wave = tid >> 5;
    const int lr   = lane & 15;     // M%16 for A-frag / N%16 for B-frag & D
    const int lh   = lane >> 4;     // lane half selects K pair (A/B) and M+8 (D)
    const int wm   = (wave >> 1) * 32;   // wave's 32x32 sub-tile
    const int wn   = (wave & 1)  * 32;

    const int m0 = blockIdx.y * 64;
    const int n0 = blockIdx.x * 64;
    const int b  = blockIdx.z;

    const long long offA = (long long)b * sAb;
    const long long offB = (long long)b * sBb;

    // ---- pipeline stages -------------------------------------------------
    auto fetchA = [&](int kc, float (&r)[8]) {
        #pragma unroll
        for (int i = 0; i < 8; ++i) {              // A: 64x16
            int idx = tid + i * 128;
            int m = idx >> 4, k = idx & 15;
            int mg = m0 + m, kg = kc + k;
            bool in = (mg < M) & (kg < K);
            long long o = in ? (offA + (long long)mg * sAm + (long long)kg * sAk) : offA;
            r[i] = A[o];
            __builtin_prefetch(&A[o + 16 * sAk], 0, 3);   // warm L2 one tile further
        }
    };
    auto fetchB = [&](int kc, float (&r)[8]) {
        #pragma unroll
        for (int i = 0; i < 8; ++i) {              // B: 16x64
            int idx = tid + i * 128;
            int k = idx >> 6, n = idx & 63;
            int kg = kc + k, ng = n0 + n;
            bool in = (kg < K) & (ng < N);
            bool lo = kg < K0;
            const float* Bp = lo ? B0 : B1;
            long long kk = lo ? (long long)kg : (long long)(kg - K0);
            long long o = in ? (offB + kk * sBk + (long long)ng * sBn) : offB;
            r[i] = Bp[o];
            __builtin_prefetch(&Bp[o + 16 * sBk], 0, 3);
        }
    };
    auto stash = [&](int kc, float (&ra)[8], float (&rb)[8]) {
        #pragma unroll
        for (int i = 0; i < 8; ++i) {
            int idx = tid + i * 128;
            int m = idx >> 4, k = idx & 15;
            bool in = ((m0 + m) < M) & ((kc + k) < K);
            Al[m * 18 + k] = in ? ra[i] : 0.0f;
        }
        #pragma unroll
        for (int i = 0; i < 8; ++i) {
            int idx = tid + i * 128;
            int k = idx >> 6, n = idx & 63;
            bool in = ((kc + k) < K) & ((n0 + n) < N);
            Bl[n * 18 + k] = in ? rb[i] : 0.0f;
        }
    };

    v8f acc[2][2] = {};
    float ra[8], rb[8], ra2[8], rb2[8];

    fetchA(0, ra);
    fetchB(0, rb);

    for (int kc = 0; kc < K; kc += 16) {
        stash(kc, ra, rb);
        __syncthreads();

        const bool more = (kc + 16) < K;
        if (more) {                    // uniform branch; loads overlap WMMAs below
            fetchA(kc + 16, ra2);
            fetchB(kc + 16, rb2);
        }

        // ---- 4 K-substeps of 4, 2x2 tiles, 16 WMMAs ----
        #pragma unroll
        for (int ks = 0; ks < 16; ks += 4) {
            const int kb = ks + 2 * lh;            // lane's K pair origin
            v2f a0 = *(const v2f*)&Al[(wm      + lr) * 18 + kb];
            v2f a1 = *(const v2f*)&Al[(wm + 16 + lr) * 18 + kb];
            v2f b0 = *(const v2f*)&Bl[(wn      + lr) * 18 + kb];
            v2f b1 = *(const v2f*)&Bl[(wn + 16 + lr) * 18 + kb];
            acc[0][0] = wmma4(a0, b0, acc[0][0]);
            acc[0][1] = wmma4(a0, b1, acc[0][1]);
            acc[1][0] = wmma4(a1, b0, acc[1][0]);
            acc[1][1] = wmma4(a1, b1, acc[1][1]);
        }
        __syncthreads();

        if (more) {
            #pragma unroll
            for (int i = 0; i < 8; ++i) { ra[i] = ra2[i]; rb[i] = rb2[i]; }
        }
    }

    // ---- store with fused bias/BN/ReLU epilogue ----
    float* Yb = Y + (long long)b * sYb;
    #pragma unroll
    for (int ti = 0; ti < 2; ++ti) {
        #pragma unroll
        for (int tj = 0; tj < 2; ++tj) {
            const int ng = n0 + wn + tj * 16 + lr;
            #pragma unroll
            for (int v = 0; v < 8; ++v) {
                const int mg = m0 + wm + ti * 16 + v + 8 * lh;   // D: M = v + 8*(lane/16)
                if (mg < M && ng < N) {
                    float s  = scale ? scale[mg] : 1.0f;
                    float sh = shift ? shift[mg] : 0.0f;
                    float val = acc[ti][tj][v] * s + sh;
                    if (relu) val = fmaxf(val, 0.0f);
                    Yb[(long long)mg * ldY + ng] = val;
                }
            }
        }
    }
}

// ---------------------------------------------------------------------------
// Row softmax, one wave32 per row (cols <= 192)
// ---------------------------------------------------------------------------
__global__ __launch_bounds__(256) void softmax_rows(float* __restrict__ P,
                                                    long long rows, int cols)
{
    const long long row = (long long)blockIdx.x * (blockDim.x >> 5) + (threadIdx.x >> 5);
    const int lane = threadIdx.x & 31;
    if (row >= rows) return;
    float* p = P + row * cols;

    float v[6];
    #pragma unroll
    for (int i = 0; i < 6; ++i) {
        int idx = lane + 32 * i;
        v[i] = (idx < cols) ? p[idx] : -3.0e38f;
    }
    float mx = -3.0e38f;
    #pragma unroll
    for (int i = 0; i < 6; ++i) mx = fmaxf(mx, v[i]);
    #pragma unroll
    for (int off = 16; off; off >>= 1) mx = fmaxf(mx, __shfl_xor(mx, off, 32));

    float sum = 0.0f;
    #pragma unroll
    for (int i = 0; i < 6; ++i) {
        int idx = lane + 32 * i;
        v[i] = (idx < cols) ? __expf(v[i] - mx) : 0.0f;
        sum += v[i];
    }
    #pragma unroll
    for (int off = 16; off; off >>= 1) sum += __shfl_xor(sum, off, 32);

    const float inv = 1.0f / sum;
    #pragma unroll
    for (int i = 0; i < 6; ++i) {
        int idx = lane + 32 * i;
        if (idx < cols) p[idx] = v[i] * inv;
    }
}

// Fold conv bias + BN into per-channel scale/shift:
//   y = (Wx + bias)*inv + (beta - mean*inv)  ->  scale = inv, shift = bias*inv + beta - mean*inv
__global__ void prep_bn(const float* __restrict__ gamma, const float* __restrict__ beta,
                        const float* __restrict__ mean, const float* __restrict__ var,
                        const float* __restrict__ bias,
                        float* __restrict__ scale, float* __restrict__ shift, int n)
{
    int i = blockIdx.x * blockDim.x + threadIdx.x;
    if (i < n) {
        float inv = gamma[i] * rsqrtf(var[i] + 1e-5f);
        scale[i] = inv;
        shift[i] = bias[i] * inv + beta[i] - mean[i] * inv;
    }
}

extern "C" void kernel_launch(void* const* d_in, const int* in_sizes, int n_in,
                              void* d_out, int out_size, void* d_ws, size_t ws_size,
                              hipStream_t stream) {
    const float* zf      = (const float*)d_in[0];
    const float* xf      = (const float*)d_in[1];
    const float* Wq      = (const float*)d_in[2];
    const float* bq      = (const float*)d_in[3];
    const float* Wsw     = (const float*)d_in[4];
    const float* bs      = (const float*)d_in[5];
    const float* Wg      = (const float*)d_in[6];
    const float* bg      = (const float*)d_in[7];
    const float* g_gamma = (const float*)d_in[8];
    const float* g_beta  = (const float*)d_in[9];
    const float* g_mean  = (const float*)d_in[10];
    const float* g_var   = (const float*)d_in[11];
    const float* Wf      = (const float*)d_in[12];
    const float* bf      = (const float*)d_in[13];
    const float* f_gamma = (const float*)d_in[14];
    const float* f_beta  = (const float*)d_in[15];
    const float* f_mean  = (const float*)d_in[16];
    const float* f_var   = (const float*)d_in[17];

    const int B = 128, C = 256, O = 256;
    const int Nz = 13 * 13;   // 169
    const int Nx = 31 * 31;   // 961
    const long long BCNx = (long long)B * C * Nx;
    const long long BCNz = (long long)B * C * Nz;
    const long long BNxNz = (long long)B * Nx * Nz;

    float* ws = (float*)d_ws;
    size_t off = 0;
    float* xf_trans = ws + off; off += (size_t)BCNx;
    float* xf_g     = ws + off; off += (size_t)BCNx;
    float* zf_trans = ws + off; off += (size_t)BCNz;
    float* zf_g     = ws + off; off += (size_t)BCNz;
    float* sim      = ws + off; off += (size_t)BNxNz;
    float* emb      = ws + off; off += (size_t)BCNx;
    float* g_scale  = ws + off; off += C;
    float* g_shift  = ws + off; off += C;
    float* f_scale  = ws + off; off += O;
    float* f_shift  = ws + off; off += O;

    prep_bn<<<dim3((C + 127) / 128), dim3(128), 0, stream>>>(
        g_gamma, g_beta, g_mean, g_var, bg, g_scale, g_shift, C);
    prep_bn<<<dim3((O + 127) / 128), dim3(128), 0, stream>>>(
        f_gamma, f_beta, f_mean, f_var, bf, f_scale, f_shift, O);

    const dim3 blk(128);
    auto gN = [](int n) { return (unsigned)((n + 63) / 64); };

    // xf_trans = Wq @ xf + bq          [B, C, Nx]
    wmma_gemm_f32<<<dim3(gN(Nx), gN(C), B), blk, 0, stream>>>(
        Wq, C, 1, 0,  xf, xf, C, Nx, 1, (long long)C * Nx,
        xf_trans, Nx, (long long)C * Nx, C, Nx, C, nullptr, bq, 0);
    // zf_trans = Ws @ zf + bs          [B, C, Nz]
    wmma_gemm_f32<<<dim3(gN(Nz), gN(C), B), blk, 0, stream>>>(
        Wsw, C, 1, 0,  zf, zf, C, Nz, 1, (long long)C * Nz,
        zf_trans, Nz, (long long)C * Nz, C, Nz, C, nullptr, bs, 0);
    // xf_g = relu(bn(Wg @ xf + bg))    [B, C, Nx]
    wmma_gemm_f32<<<dim3(gN(Nx), gN(C), B), blk, 0, stream>>>(
        Wg, C, 1, 0,  xf, xf, C, Nx, 1, (long long)C * Nx,
        xf_g, Nx, (long long)C * Nx, C, Nx, C, g_scale, g_shift, 1);
    // zf_g = relu(bn(Wg @ zf + bg))    [B, C, Nz]
    wmma_gemm_f32<<<dim3(gN(Nz), gN(C), B), blk, 0, stream>>>(
        Wg, C, 1, 0,  zf, zf, C, Nz, 1, (long long)C * Nz,
        zf_g, Nz, (long long)C * Nz, C, Nz, C, g_scale, g_shift, 1);
    // sim[b,n,m] = sum_c xf_trans[b,c,n] * zf_trans[b,c,m]   [B, Nx, Nz]
    wmma_gemm_f32<<<dim3(gN(Nz), gN(Nx), B), blk, 0, stream>>>(
        xf_trans, 1, Nx, (long long)C * Nx,
        zf_trans, zf_trans, C, Nz, 1, (long long)C * Nz,
        sim, Nz, (long long)Nx * Nz, Nx, Nz, C, nullptr, nullptr, 0);
    // softmax over Nz
    {
        long long rows = (long long)B * Nx;
        unsigned grid = (unsigned)((rows + 7) / 8);
        softmax_rows<<<dim3(grid), dim3(256), 0, stream>>>(sim, rows, Nz);
    }
    // emb[b,c,n] = sum_m zf_g[b,c,m] * sim[b,n,m]            [B, C, Nx]
    wmma_gemm_f32<<<dim3(gN(Nx), gN(C), B), blk, 0, stream>>>(
        zf_g, Nz, 1, (long long)C * Nz,
        sim, sim, Nz, 1, Nz, (long long)Nx * Nz,
        emb, Nx, (long long)C * Nx, C, Nx, Nz, nullptr, nullptr, 0);
    // out = relu(bn(Wf @ concat(emb, xf_g) + bf))            [B, O, Nx]
    wmma_gemm_f32<<<dim3(gN(Nx), gN(O), B), blk, 0, stream>>>(
        Wf, 2 * C, 1, 0,
        emb, xf_g, C, Nx, 1, (long long)C * Nx,
        (float*)d_out, Nx, (long long)O * Nx, O, Nx, 2 * C, f_scale, f_shift, 1);
}